// SparseGCN_58411555225966
// MI455X (gfx1250) — compile-verified
//
#include <hip/hip_runtime.h>

typedef __attribute__((ext_vector_type(2))) float v2f;
typedef __attribute__((ext_vector_type(8))) float v8f;

#define FDIM 128
#define LDSTRIDE 132   // 132 mod 64 == 4 -> conflict-free column reads

// ---------------- init: deg=1 (self loop), Acc=0, s=0 ----------------
__global__ void gcn_init(float* __restrict__ deg, float* __restrict__ acc,
                         float* __restrict__ s, int n) {
    size_t i = (size_t)blockIdx.x * blockDim.x + threadIdx.x;
    size_t total = (size_t)n * FDIM;
    if (i < total) acc[i] = 0.0f;
    if (i < (size_t)n) deg[i] = 1.0f;
    if (i < FDIM) s[i] = 0.0f;
}

// ---------------- degree accumulation over edges ----------------
__global__ void gcn_deg(const int* __restrict__ row, float* __restrict__ deg, int E) {
    int e = blockIdx.x * blockDim.x + threadIdx.x;
    if (e < E) atomicAdd(&deg[row[e]], 1.0f);
}

// ------------- dis = rsqrt(deg); vcol init = self_w = dis^2 -------------
__global__ void gcn_dis(float* __restrict__ deg_dis, float* __restrict__ vcol, int n) {
    int i = blockIdx.x * blockDim.x + threadIdx.x;
    if (i < n) {
        float d = __frsqrt_rn(deg_dis[i]);   // deg >= 1 always
        deg_dis[i] = d;                      // in place: now holds d_inv_sqrt
        vcol[i]    = d * d;                  // self weight
    }
}

// ------------- vcol[c] += edge_w  (the 1^T A  part of the readout) -------------
__global__ void gcn_vcol(const int* __restrict__ row, const int* __restrict__ col,
                         const float* __restrict__ dis, float* __restrict__ vcol, int E) {
    int e = blockIdx.x * blockDim.x + threadIdx.x;
    if (e < E) {
        int r = row[e], c = col[e];
        atomicAdd(&vcol[c], dis[r] * dis[c]);
    }
}

// ---------------- Y = F @ W1 via v_wmma_f32_16x16x4_f32 ----------------
// block = 256 threads = 8 waves; block computes a 16x128 strip of Y.
// wave w computes the 16x16 tile at columns [16w, 16w+16).
__global__ void gcn_gemm(const float* __restrict__ F, const float* __restrict__ W1,
                         float* __restrict__ Y, int n) {
    __shared__ float As[16 * LDSTRIDE];
    const int row0 = blockIdx.x * 16;
    const int tid  = threadIdx.x;

    // stage the 16x128 A strip (coalesced)
    for (int i = tid; i < 16 * FDIM; i += 256) {
        int r = i >> 7, c = i & (FDIM - 1);
        float v = (row0 + r < n) ? F[(size_t)(row0 + r) * FDIM + c] : 0.0f;
        As[r * LDSTRIDE + c] = v;
    }
    __syncthreads();

    const int wave = tid >> 5;
    const int lane = tid & 31;
    const int col0 = wave * 16;
    const int m    = lane & 15;          // row within tile (A) / col within tile (B,D)
    const int kb   = (lane >> 4) * 2;    // lanes 16-31 carry K+2,K+3

    v8f acc = {};
    for (int k = 0; k < FDIM; k += 4) {
        v2f a, b;
        a.x = As[m * LDSTRIDE + k + kb];
        a.y = As[m * LDSTRIDE + k + kb + 1];
        b.x = W1[(size_t)(k + kb)     * FDIM + col0 + m];
        b.y = W1[(size_t)(k + kb + 1) * FDIM + col0 + m];
        acc = __builtin_amdgcn_wmma_f32_16x16x4_f32(
                  false, a, false, b, (short)0, acc, false, false);
    }

    // D layout: VGPR j -> row j (lanes 0-15) / row j+8 (lanes 16-31), col = lane&15
    const int rbase = row0 + ((lane >> 4) ? 8 : 0);
    const int ccol  = col0 + m;
    for (int j = 0; j < 8; ++j) {
        int r = rbase + j;
        if (r < n) Y[(size_t)r * FDIM + ccol] = acc[j];
    }
}

// ---------- SpMM scatter: Acc[row] += edge_w * Y[col], one wave per edge ----------
__global__ void gcn_spmm(const int* __restrict__ row, const int* __restrict__ col,
                         const float* __restrict__ dis, const float* __restrict__ Y,
                         float* __restrict__ acc, int E) {
    int gwarp = (blockIdx.x * blockDim.x + threadIdx.x) >> 5;
    int lane  = threadIdx.x & 31;
    if (gwarp >= E) return;
    int r = row[gwarp], c = col[gwarp];
    float w = dis[r] * dis[c];
    float4 v = ((const float4*)(Y + (size_t)c * FDIM))[lane];  // 512B coalesced, L2-resident
    float* dst = acc + (size_t)r * FDIM + lane * 4;
    atomicAdd(dst + 0, w * v.x);
    atomicAdd(dst + 1, w * v.y);
    atomicAdd(dst + 2, w * v.z);
    atomicAdd(dst + 3, w * v.w);
}

// ----- s[f] = sum_n vcol[n] * relu(Acc[n,f] + dis[n]^2*Y[n,f] + b1[f]) -----
#define NODES_PER_BLOCK 128
__global__ void gcn_reduce(const float* __restrict__ acc, const float* __restrict__ Y,
                           const float* __restrict__ dis, const float* __restrict__ vcol,
                           const float* __restrict__ b1, float* __restrict__ s, int n) {
    int f = threadIdx.x;                       // 0..127, one feature per lane
    float bf = b1[f];
    int n0 = blockIdx.x * NODES_PER_BLOCK;
    int n1 = min(n0 + NODES_PER_BLOCK, n);
    float sum = 0.0f;
    for (int node = n0; node < n1; ++node) {
        float d = dis[node];
        float z = acc[(size_t)node * FDIM + f] + d * d * Y[(size_t)node * FDIM + f] + bf;
        z = fmaxf(z, 0.0f);
        sum += vcol[node] * z;
    }
    atomicAdd(&s[f], sum);
}

// ---------------- out = (s/N) @ W2 + b2 ----------------
__global__ void gcn_out(const float* __restrict__ s, const float* __restrict__ W2,
                        const float* __restrict__ b2, float* __restrict__ out, float invN) {
    int c = threadIdx.x;
    if (c < 40) {
        float a = b2[c];
        for (int f = 0; f < FDIM; ++f) a += (s[f] * invN) * W2[f * 40 + c];
        out[c] = a;
    }
}

extern "C" void kernel_launch(void* const* d_in, const int* in_sizes, int n_in,
                              void* d_out, int out_size, void* d_ws, size_t ws_size,
                              hipStream_t stream) {
    const int  E = in_sizes[0] / 2;
    const int  N = in_sizes[1] / FDIM;
    const int* row = (const int*)d_in[0];       // edge_index[0], JAX x64-off -> int32
    const int* col = row + E;                   // edge_index[1]
    const float* F  = (const float*)d_in[1];
    const float* W1 = (const float*)d_in[2];
    const float* b1 = (const float*)d_in[3];
    const float* W2 = (const float*)d_in[4];
    const float* b2 = (const float*)d_in[5];
    float* out = (float*)d_out;

    // workspace layout (floats): Y[N*128] | Acc[N*128] | dis[N] | vcol[N] | s[128]
    float* Y    = (float*)d_ws;
    float* Acc  = Y   + (size_t)N * FDIM;
    float* dis  = Acc + (size_t)N * FDIM;       // holds deg, then d_inv_sqrt in place
    float* vcol = dis + N;
    float* s    = vcol + N;

    const size_t total = (size_t)N * FDIM;
    gcn_init<<<(unsigned)((total + 255) / 256), 256, 0, stream>>>(dis, Acc, s, N);
    gcn_deg <<<(E + 255) / 256, 256, 0, stream>>>(row, dis, E);
    gcn_dis <<<(N + 255) / 256, 256, 0, stream>>>(dis, vcol, N);
    gcn_gemm<<<(N + 15) / 16, 256, 0, stream>>>(F, W1, Y, N);
    gcn_vcol<<<(E + 255) / 256, 256, 0, stream>>>(row, col, dis, vcol, E);
    gcn_spmm<<<(unsigned)(((size_t)E * 32 + 255) / 256), 256, 0, stream>>>(row, col, dis, Y, Acc, E);
    gcn_reduce<<<(N + NODES_PER_BLOCK - 1) / NODES_PER_BLOCK, FDIM, 0, stream>>>(Acc, Y, dis, vcol, b1, s, N);
    gcn_out<<<1, 64, 0, stream>>>(s, W2, b2, out, 1.0f / (float)N);
}